// CrossEntropy2d_self_supervised_39668317946483
// MI455X (gfx1250) — compile-verified
//
#include <hip/hip_runtime.h>
#include <hip/hip_bf16.h>
#include <math.h>

typedef __attribute__((ext_vector_type(2))) float v2f;
typedef __attribute__((ext_vector_type(8))) float v8f;

#define N_      8
#define CF_     32
#define NC_     2
#define H_      512
#define W_      512
#define HW_     (H_ * W_)
#define K_      16384
#define GAMMA_  0.9f
#define BETA_   0.5f

// ---- workspace layout -------------------------------------------------------
// [0 .. BITS_WORDS)            : bg membership bitmap (1 bit / pixel / item)
// float section F[...]:
#define BITS_WORDS ((N_ * HW_) / 32)        // 65536 words = 256 KB
#define F_C0SUM 0        // 256: sum over bg columns, per (n,c)
#define F_C1SUM 256      // 256: sum over fg columns
#define F_R0SQ  512      // 256: sum of squares over bg columns
#define F_R1SQ  768      // 256: sum of squares over fg columns
#define F_C0M   1024     // 256: c0 mean
#define F_C1M   1280     // 256: c1 mean
#define F_N0    1536     // 8
#define F_N1    1544     // 8
#define F_FNUM  1552     // 8
#define F_FDEN  1560     // 8
#define F_SUP   1568     // 1: raw sum of logp[target] over sup set (all n)
#define F_SELFN 1569     // 8: per-n masked CE sum
#define F_SELFC 1577     // 8: per-n mask count
#define F_TOTAL 1585

// ---- kernel 1: zero the workspace ------------------------------------------
__global__ void k_zero(unsigned* ws, int nwords) {
    for (int i = blockIdx.x * blockDim.x + threadIdx.x; i < nwords;
         i += gridDim.x * blockDim.x)
        ws[i] = 0u;
}

// ---- kernel 2: scatter bg indices into bitmap ------------------------------
__global__ void k_scatter_bg(const int* __restrict__ bg, unsigned* __restrict__ bits) {
    int i = blockIdx.x * blockDim.x + threadIdx.x;
    if (i >= N_ * K_) return;
    int n   = i / K_;
    int pos = n * HW_ + bg[i];
    atomicOr(&bits[pos >> 5], 1u << (pos & 31));
}

// ---- kernel 3: per-(n,c) gather reductions ---------------------------------
// block b = n*32+c; 256 threads stride over K indices of both fg and bg sets.
__global__ void __launch_bounds__(256) k_centroid(const float* __restrict__ act,
                                                  const int* __restrict__ fg,
                                                  const int* __restrict__ bg,
                                                  float* __restrict__ F) {
    int b = blockIdx.x;                 // 0..255
    int n = b >> 5;
    int c = b & 31;
    const float* plane = act + ((size_t)n * CF_ + c) * HW_;
    const int*   bgp   = bg + (size_t)n * K_;
    const int*   fgp   = fg + (size_t)n * K_;
    float s0 = 0.f, q0 = 0.f, s1 = 0.f, q1 = 0.f;
    for (int k = threadIdx.x; k < K_; k += blockDim.x) {
        float v = plane[bgp[k]];
        s0 += v;  q0 = fmaf(v, v, q0);
        float w = plane[fgp[k]];
        s1 += w;  q1 = fmaf(w, w, q1);
    }
    __shared__ float4 red[256];
    red[threadIdx.x] = make_float4(s0, q0, s1, q1);
    __syncthreads();
    for (int off = 128; off > 0; off >>= 1) {
        if (threadIdx.x < off) {
            float4 a = red[threadIdx.x], bb = red[threadIdx.x + off];
            red[threadIdx.x] = make_float4(a.x + bb.x, a.y + bb.y,
                                           a.z + bb.z, a.w + bb.w);
        }
        __syncthreads();
    }
    if (threadIdx.x == 0) {
        float4 r = red[0];
        F[F_C0SUM + b] = r.x;
        F[F_R0SQ  + b] = r.y;
        F[F_C1SUM + b] = r.z;
        F[F_R1SQ  + b] = r.w;
    }
}

// ---- kernel 4: finalize centroids + fisher terms (one wave32 per n) --------
__global__ void k_finalize(float* __restrict__ F) {
    int n = blockIdx.x;
    int c = threadIdx.x;                 // 0..31
    int b = n * 32 + c;
    float c0 = F[F_C0SUM + b] * (1.f / (float)K_);
    float c1 = F[F_C1SUM + b] * (1.f / (float)K_);
    F[F_C0M + b] = c0;
    F[F_C1M + b] = c1;
    float a00 = c0 * c0;
    float a11 = c1 * c1;
    float a01 = c0 * c1;
    float d0  = c0 * c0 * F[F_R0SQ + b];
    float d1  = c1 * c1 * F[F_R1SQ + b];
    #pragma unroll
    for (int off = 16; off > 0; off >>= 1) {
        a00 += __shfl_xor(a00, off, 32);
        a11 += __shfl_xor(a11, off, 32);
        a01 += __shfl_xor(a01, off, 32);
        d0  += __shfl_xor(d0,  off, 32);
        d1  += __shfl_xor(d1,  off, 32);
    }
    if (c == 0) {
        float n0 = sqrtf(a00), n1 = sqrtf(a11);
        F[F_N0 + n] = n0;
        F[F_N1 + n] = n1;
        F[F_FNUM + n] = a01 / (n0 * n1);
        // sum(sym0) = K*|c0|^2 / ||c0 o rowNorm(bg_feats)||  (analytically reduced)
        F[F_FDEN + n] = (float)K_ * (a00 * rsqrtf(d0) + a11 * rsqrtf(d1));
    }
}

// ---- kernel 5: main full-image pass with WMMA ------------------------------
// grid = (HW/128, N); block = 256 = 8 waves; each wave handles 16 pixels.
// A (16x4 f32) rows: row0=c0 chunk, row1=c1 chunk, rows 2..15 = 0.
// B (4x16 f32): lanes 0-15 hold K=0,1 for pixel=col; lanes 16-31 hold K=2,3.
// A is loop-invariant: preloaded to registers. All 16 B loads issued as one
// clause, then 8 chained v_wmma_f32_16x16x4_f32.
__global__ void __launch_bounds__(256) k_main(const float* __restrict__ act,
                                              const float* __restrict__ pred,
                                              const int* __restrict__ tgt,
                                              const unsigned* __restrict__ bits,
                                              float* __restrict__ F) {
    const int n    = blockIdx.y;
    const int wave = threadIdx.x >> 5;
    const int lane = threadIdx.x & 31;
    const int half = lane >> 4;
    const int col  = lane & 15;
    const int base = blockIdx.x * 128 + wave * 16;
    const int pix  = base + col;
    const float* actn = act + (size_t)n * CF_ * HW_;
    const float* c0p  = F + F_C0M + n * 32;
    const float* c1p  = F + F_C1M + n * 32;

    // ---- hoisted A operand: every lane loads the selected row (no divergent
    //      pointer-null path), then zero-selects rows 2..15.
    const float* rowp = (col == 1) ? c1p : c0p;
    const bool   useA = (col < 2);
    v2f A[8];
    #pragma unroll
    for (int kk = 0; kk < 8; ++kk) {
        int ch = 4 * kk + 2 * half;
        float ax = rowp[ch];
        float ay = rowp[ch + 1];
        A[kk].x = useA ? ax : 0.f;
        A[kk].y = useA ? ay : 0.f;
    }
    const float n0v = F[F_N0 + n];
    const float n1v = F[F_N1 + n];

    // ---- issue all 16 B loads (16 channels / wave-half) as one clause
    float bx[8], by[8];
    #pragma unroll
    for (int kk = 0; kk < 8; ++kk) {
        int ch = 4 * kk + 2 * half;
        bx[kk] = actn[(size_t)ch * HW_ + pix];
        by[kk] = actn[(size_t)(ch + 1) * HW_ + pix];
    }

    // |a|^2 partials while loads land
    float nrm = 0.f;
    #pragma unroll
    for (int kk = 0; kk < 8; ++kk) {
        nrm = fmaf(bx[kk], bx[kk], nrm);
        nrm = fmaf(by[kk], by[kk], nrm);
    }

    // ---- 8 chained WMMAs: D = A*B + C over K=32 in steps of 4
    v8f acc = {};
    #pragma unroll
    for (int kk = 0; kk < 8; ++kk) {
        v2f B = {bx[kk], by[kk]};
        acc = __builtin_amdgcn_wmma_f32_16x16x4_f32(
                  false, A[kk], false, B, (short)0, acc, false, false);
    }

    // combine half-wave |a|^2 partials (channels {4k,4k+1} + {4k+2,4k+3})
    nrm += __shfl_xor(nrm, 16, 32);

    if (half == 0) {                       // lanes 0..15 own the 16 pixels
        float s0   = acc[0];               // D row 0 = c0 . a
        float s1   = acc[1];               // D row 1 = c1 . a
        float invn = rsqrtf(nrm);
        float sim0 = s0 * invn / n0v;
        float sim1 = s1 * invn / n1v;
        int pseudo = (sim1 > GAMMA_) ? 1 : ((sim0 > GAMMA_) ? 0 : 2);

        const float* pn = pred + (size_t)n * NC_ * HW_;
        float p0 = pn[pix];
        float p1 = pn[HW_ + pix];
        float mx  = fmaxf(p0, p1);
        float lse = mx + __logf(__expf(p0 - mx) + __expf(p1 - mx));
        float lp0 = p0 - lse, lp1 = p1 - lse;

        int gp = n * HW_ + pix;
        int tg = tgt[gp];
        bool inbg = (bits[gp >> 5] >> (gp & 31)) & 1u;

        float supv  = (tg == 1) ? lp1 : (inbg ? lp0 : 0.f);
        bool  msk   = (tg == 0) && !inbg && (pseudo != 2);
        float selfv = msk ? -((pseudo == 1) ? lp1 : lp0) : 0.f;
        float cntv  = msk ? 1.f : 0.f;

        #pragma unroll
        for (int off = 8; off > 0; off >>= 1) {
            supv  += __shfl_down(supv,  off, 32);
            selfv += __shfl_down(selfv, off, 32);
            cntv  += __shfl_down(cntv,  off, 32);
        }
        if (col == 0) {
            atomicAdd(&F[F_SUP],       supv);
            atomicAdd(&F[F_SELFN + n], selfv);
            atomicAdd(&F[F_SELFC + n], cntv);
        }
    }
}

// ---- kernel 6: scalar combine ----------------------------------------------
__global__ void k_final(const float* __restrict__ F,
                        const int* __restrict__ tP,  const int* __restrict__ T1P,
                        const int* __restrict__ T2P, const int* __restrict__ lamP,
                        float* __restrict__ out) {
    if (threadIdx.x != 0 || blockIdx.x != 0) return;
    float sup = -F[F_SUP] / (float)(2 * K_);   // all items share |sup_idx| = 2K
    float selfTot = 0.f, fn = 0.f, fd = 0.f;
    for (int n = 0; n < N_; ++n) {
        selfTot += F[F_SELFN + n] / F[F_SELFC + n];   // 0/0 -> NaN matches jnp
        fn += F[F_FNUM + n];
        fd += F[F_FDEN + n];
    }
    float fisher = fn / fd;
    int t = *tP, T1 = *T1P, T2 = *T2P, lam = *lamP;
    float loss;
    if (t < T1) {
        loss = sup;
    } else {
        float alpha = (t < T2) ? (float)(t - T1) * (float)lam / (float)(T2 - T1)
                               : (float)lam;
        float l = sup + alpha * selfTot + BETA_ * fisher;
        bool bad = (selfTot != selfTot) || (fisher != fisher);
        loss = bad ? sup : l;
    }
    out[0] = loss;
}

// ---- launcher ---------------------------------------------------------------
extern "C" void kernel_launch(void* const* d_in, const int* in_sizes, int n_in,
                              void* d_out, int out_size, void* d_ws, size_t ws_size,
                              hipStream_t stream) {
    const float* act  = (const float*)d_in[0];   // (N, CF, H, W)
    const float* pred = (const float*)d_in[1];   // (N, NC, H, W)
    const int*   tgt  = (const int*)d_in[2];     // (N, H, W)
    // d_in[3] = full_target (unused by reference)
    const int*   fg   = (const int*)d_in[4];     // (N, K)
    const int*   bg   = (const int*)d_in[5];     // (N, K)
    const int*   tP   = (const int*)d_in[6];
    const int*   T1P  = (const int*)d_in[7];
    const int*   T2P  = (const int*)d_in[8];
    const int*   lamP = (const int*)d_in[9];

    unsigned* bits = (unsigned*)d_ws;
    float*    F    = (float*)d_ws + BITS_WORDS;
    const int nwords = BITS_WORDS + F_TOTAL;

    k_zero<<<256, 256, 0, stream>>>((unsigned*)d_ws, nwords);
    k_scatter_bg<<<(N_ * K_ + 255) / 256, 256, 0, stream>>>(bg, bits);
    k_centroid<<<N_ * CF_, 256, 0, stream>>>(act, fg, bg, F);
    k_finalize<<<N_, 32, 0, stream>>>(F);
    dim3 grid(HW_ / 128, N_);
    k_main<<<grid, 256, 0, stream>>>(act, pred, tgt, bits, F);
    k_final<<<1, 1, 0, stream>>>(F, tP, T1P, T2P, lamP, (float*)d_out);
}